// NodeModule_46703474377027
// MI455X (gfx1250) — compile-verified
//
#include <hip/hip_runtime.h>

typedef float v2f __attribute__((ext_vector_type(2)));
typedef float v8f __attribute__((ext_vector_type(8)));

#define TPAD   1040          // 65 tiles * 16
#define NROWS  1025          // K*T*2 + 1
#define NTILE  65
#define P2ROWS 1050625       // 1025*1025
#define ROWS_OUT 1051651     // P2ROWS + 1025 + 1
#define SOUT   29446228      // ROWS_OUT * 28 floats per returned tensor

// workspace float offsets
#define WS_WU   0            // weighted under, planar [(v*4+i)*TPAD + t], 16640 floats
#define WS_WO   16640
#define WS_U    33280        // raw under
#define WS_O    49920
#define WS_LUND 66560        // per-row lower bound of under
#define WS_UOVR 67600        // per-row upper bound of over
#define WS_SCAL 68640        // l,u,a_u,b_u,c_u,a_o,b_o,c_o

__device__ __forceinline__ int imax(int a, int b) { return a > b ? a : b; }

// Store 7 contiguous floats at p = base + R*28 + v*7 (R*112 bytes is 16B-aligned),
// using the widest aligned stores for each v-phase. v must be compile-time.
__device__ __forceinline__ void store7(float* __restrict__ p, int v,
                                       float d0, float d1, float d2, float d3,
                                       float d4, float d5, float d6) {
  if (v == 0) {        // byte 0: 16-al
    float4 q = {d0,d1,d2,d3}; *(float4*)(p) = q;
    float2 w = {d4,d5};       *(float2*)(p+4) = w;
    p[6] = d6;
  } else if (v == 1) { // byte 28: +1 -> 16-al, +5 -> 8-al
    p[0] = d0;
    float4 q = {d1,d2,d3,d4}; *(float4*)(p+1) = q;
    float2 w = {d5,d6};       *(float2*)(p+5) = w;
  } else if (v == 2) { // byte 56: 8-al, +2 -> 16-al
    float2 w = {d0,d1};       *(float2*)(p) = w;
    float4 q = {d2,d3,d4,d5}; *(float4*)(p+2) = q;
    p[6] = d6;
  } else {             // byte 84: +1 -> 8-al, +3 -> 16-al
    p[0] = d0;
    float2 w = {d1,d2};       *(float2*)(p+1) = w;
    float4 q = {d3,d4,d5,d6}; *(float4*)(p+3) = q;
  }
}

// ---------------------------------------------------------------------------
// Stage 1: build under/over polynomials (raw + C3-weighted planar), intervals.
// ---------------------------------------------------------------------------
__global__ void k_prep(const float* __restrict__ iu, const float* __restrict__ io,
                       const float* __restrict__ posw, const float* __restrict__ negw,
                       const float* __restrict__ bias, float* __restrict__ ws) {
  int t = blockIdx.x * blockDim.x + threadIdx.x;
  if (t >= TPAD) return;
  float* wuP = ws + WS_WU; float* woP = ws + WS_WO;
  float* uP  = ws + WS_U;  float* oP  = ws + WS_O;
  const float C3[4] = {1.f, 3.f, 3.f, 1.f};

  if (t >= NROWS) {  // zero padding rows
    for (int v = 0; v < 4; ++v)
      for (int i = 0; i < 4; ++i) {
        int pi = (v*4+i)*TPAD + t;
        wuP[pi] = 0.f; woP[pi] = 0.f; uP[pi] = 0.f; oP[pi] = 0.f;
      }
    ws[WS_LUND + t] = 0.f; ws[WS_UOVR + t] = 0.f;
    return;
  }

  float uu[4][4], oo[4][4];
  if (t == 1024) {               // bias constant row
    float b = bias[0];
    for (int v = 0; v < 4; ++v)
      for (int i = 0; i < 4; ++i) { float c = (v==0) ? b : 1.f; uu[v][i] = c; oo[v][i] = c; }
  } else {
    bool first = t < 512;
    int k  = first ? (t >> 4) : ((t - 512) >> 4);
    int tt = t & 15;
    const float* src = first ? io : iu;       // under: [io*neg, iu*pos]; over: [io*pos, iu*neg]
    float wU = first ? negw[k] : posw[k];
    float wO = first ? posw[k] : negw[k];
    int base = k*256 + tt*16;                 // (K, T*n, L) row-major
    for (int v = 0; v < 4; ++v)
      for (int i = 0; i < 4; ++i) {
        float x = src[base + v*4 + i];
        uu[v][i] = (v==0) ? x*wU : x;
        oo[v][i] = (v==0) ? x*wO : x;
      }
  }

  // interval bounds: per-var min/max of Bernstein coeffs, then interval product
  float luv[4], huv[4], lov[4], hov[4];
  for (int v = 0; v < 4; ++v) {
    float lu = uu[v][0], hu = uu[v][0], lo = oo[v][0], ho = oo[v][0];
    for (int i = 1; i < 4; ++i) {
      lu = fminf(lu, uu[v][i]); hu = fmaxf(hu, uu[v][i]);
      lo = fminf(lo, oo[v][i]); ho = fmaxf(ho, oo[v][i]);
    }
    luv[v] = lu; huv[v] = hu; lov[v] = lo; hov[v] = ho;
  }
  float Lc = luv[0], Uc = huv[0];
  for (int v = 1; v < 4; ++v) {
    float c0 = Lc*luv[v], c1 = Lc*huv[v], c2 = Uc*luv[v], c3 = Uc*huv[v];
    Lc = fminf(fminf(c0,c1), fminf(c2,c3));
    Uc = fmaxf(fmaxf(c0,c1), fmaxf(c2,c3));
  }
  ws[WS_LUND + t] = Lc;
  float Lo = lov[0], Uo = hov[0];
  for (int v = 1; v < 4; ++v) {
    float c0 = Lo*lov[v], c1 = Lo*hov[v], c2 = Uo*lov[v], c3 = Uo*hov[v];
    Lo = fminf(fminf(c0,c1), fminf(c2,c3));
    Uo = fmaxf(fmaxf(c0,c1), fmaxf(c2,c3));
  }
  ws[WS_UOVR + t] = Uo;

  for (int v = 0; v < 4; ++v)
    for (int i = 0; i < 4; ++i) {
      int pi = (v*4+i)*TPAD + t;
      uP[pi] = uu[v][i];  oP[pi] = oo[v][i];
      wuP[pi] = uu[v][i]*C3[i];  woP[pi] = oo[v][i]*C3[i];
    }
}

// ---------------------------------------------------------------------------
// Stage 2: reduce bounds -> l, u -> quad coefficients.
// ---------------------------------------------------------------------------
__global__ void k_reduce(float* __restrict__ ws) {
  __shared__ float sL[256], sU[256];
  int tid = threadIdx.x;
  float l = 0.f, u = 0.f;
  for (int t = tid; t < NROWS; t += 256) { l += ws[WS_LUND + t]; u += ws[WS_UOVR + t]; }
  sL[tid] = l; sU[tid] = u; __syncthreads();
  for (int s = 128; s > 0; s >>= 1) {
    if (tid < s) { sL[tid] += sL[tid+s]; sU[tid] += sU[tid+s]; }
    __syncthreads();
  }
  if (tid == 0) {
    float L = sL[0], U = sU[0];
    float inv = 1.f/(U - L), inv2 = inv*inv;
    float* sc = ws + WS_SCAL;
    sc[0] = L;  sc[1] = U;
    sc[2] = inv;        sc[3] = -L*inv;         sc[4] = 0.f;        // cu
    sc[5] = U*inv2;     sc[6] = -2.f*U*L*inv2;  sc[7] = U*L*L*inv2; // co
  }
}

// ---------------------------------------------------------------------------
// Stage 3: the big p^2 pair term via V_WMMA_F32_16X16X4_F32.
// For each (v,k): D[a,b] = sum_i A[a,i]*Bshift_k[i,b], K=4 contraction.
// a-coef folded into A (v=0), 1/C(6,k) folded into B.
// One wave per 16x16 (a,b) tile.  EDGE=false (interior) has no store guards:
// straight-line clause-able b128/b96 stores, no exec save/restore.
// ---------------------------------------------------------------------------
template<bool EDGE>
__global__ __launch_bounds__(32) void k_pairs_t(const float* __restrict__ ws,
                                                float* __restrict__ out,
                                                int aTileBase, int bTileBase) {
  const int lane = threadIdx.x;
  const int h = lane >> 4;          // half-wave: selects kappa pair {2h, 2h+1}
  const int r = lane & 15;          // row/col within tile
  const int a0 = (aTileBase + blockIdx.x) * 16;
  const int b0 = (bTileBase + blockIdx.y) * 16;
  const float invC6[7] = {1.f, 1.f/6.f, 1.f/15.f, 1.f/20.f, 1.f/15.f, 1.f/6.f, 1.f};
  const float* sc = ws + WS_SCAL;
  const int n = b0 + r;
  const bool nok = EDGE ? (n < NROWS) : true;

  for (int pass = 0; pass < 2; ++pass) {       // 0: ru (under,a_u)  1: ro (over,a_o)
    const float* W = ws + (pass ? WS_WO : WS_WU);
    const float acoef = pass ? sc[5] : sc[2];
    float* base = out + (size_t)pass * (size_t)SOUT;

    #pragma unroll
    for (int v = 0; v < 4; ++v) {
      // A fragment: lane r holds M=r; elem x -> kappa = 2h+x
      float av0 = W[(v*4 + 2*h    )*TPAD + a0 + r];
      float av1 = W[(v*4 + 2*h + 1)*TPAD + a0 + r];
      if (v == 0) { av0 *= acoef; av1 *= acoef; }   // compile-time after unroll
      v2f A; A[0] = av0; A[1] = av1;

      // B source coefficients for column n = b0+r (registers, constant-indexed)
      float wb[4];
      wb[0] = W[(v*4+0)*TPAD + n];
      wb[1] = W[(v*4+1)*TPAD + n];
      wb[2] = W[(v*4+2)*TPAD + n];
      wb[3] = W[(v*4+3)*TPAD + n];

      v8f D[7];
      #pragma unroll
      for (int k = 0; k < 7; ++k) {
        v2f B;
        #pragma unroll
        for (int x = 0; x < 2; ++x) {
          // j = k - kappa, kappa = 2h+x.  Both candidates are compile-time
          // register selects; one per-lane cndmask on h (branchless).
          const int j0 = k - x;          // h == 0 case
          const int j1 = k - x - 2;      // h == 1 case
          float c0 = (j0 >= 0 && j0 < 4) ? wb[j0] : 0.f;
          float c1 = (j1 >= 0 && j1 < 4) ? wb[j1] : 0.f;
          float bv = (h == 0) ? c0 : c1;
          B[x] = bv * invC6[k];
        }
        v8f C = {};
        D[k] = __builtin_amdgcn_wmma_f32_16x16x4_f32(
            /*neg_a=*/false, A, /*neg_b=*/false, B,
            /*c_mod=*/(short)0, C, /*reuse_a=*/false, /*reuse_b=*/false);
      }

      // store: lane (h,r), VGPR g -> element (M=g+8h, N=r); 7 contiguous floats
      #pragma unroll
      for (int g = 0; g < 8; ++g) {
        int m = a0 + g + 8*h;
        bool ok = EDGE ? (nok && m < NROWS) : true;
        if (ok) {
          float* p = base + ((size_t)m * NROWS + (size_t)n) * 28 + v*7;
          store7(p, v, D[0][g], D[1][g], D[2][g], D[3][g], D[4][g], D[5][g], D[6][g]);
        }
      }
    }
  }
}

// ---------------------------------------------------------------------------
// Stage 4: b*p degree-elevated rows, constant row, and 2*deg outputs.
// ---------------------------------------------------------------------------
__global__ void k_tail(const float* __restrict__ ws,
                       const int* __restrict__ udeg, const int* __restrict__ odeg,
                       float* __restrict__ out) {
  int t = blockIdx.x * blockDim.x + threadIdx.x;
  if (t > 1026) return;
  const float* sc = ws + WS_SCAL;
  if (t < NROWS) {
    const float E[4][7] = {   // elevation deg3 -> deg6
      {1.f, 0.5f, 0.2f, 0.05f, 0.f,  0.f,  0.f},
      {0.f, 0.5f, 0.6f, 0.45f, 0.2f, 0.f,  0.f},
      {0.f, 0.f,  0.2f, 0.45f, 0.6f, 0.5f, 0.f},
      {0.f, 0.f,  0.f,  0.05f, 0.2f, 0.5f, 1.f}};
    for (int pass = 0; pass < 2; ++pass) {
      const float* P = ws + (pass ? WS_O : WS_U);
      float bcoef = pass ? sc[6] : sc[3];
      float* base = out + (size_t)pass*(size_t)SOUT + ((size_t)P2ROWS + (size_t)t)*28;
      #pragma unroll
      for (int v = 0; v < 4; ++v) {
        float s = (v==0) ? bcoef : 1.f;
        float p0 = s*P[(v*4+0)*TPAD+t], p1 = s*P[(v*4+1)*TPAD+t];
        float p2 = s*P[(v*4+2)*TPAD+t], p3 = s*P[(v*4+3)*TPAD+t];
        float d[7];
        #pragma unroll
        for (int k = 0; k < 7; ++k)
          d[k] = p0*E[0][k] + p1*E[1][k] + p2*E[2][k] + p3*E[3][k];
        store7(base + v*7, v, d[0], d[1], d[2], d[3], d[4], d[5], d[6]);
      }
    }
  } else if (t == NROWS) {  // constant term row (last row)
    for (int pass = 0; pass < 2; ++pass) {
      float c = pass ? sc[7] : sc[4];
      float* base = out + (size_t)pass*(size_t)SOUT + ((size_t)(ROWS_OUT-1))*28;
      #pragma unroll
      for (int v = 0; v < 4; ++v) {
        float val = (v==0) ? c : 1.f;
        store7(base + v*7, v, val, val, val, val, val, val, val);
      }
    }
  } else {                  // 2*deg, written twice
    for (int v = 0; v < 4; ++v) {
      int mx = -2147483647;
      for (int k = 0; k < 32; ++k) mx = imax(mx, imax(udeg[k*4+v], odeg[k*4+v]));
      float dv = 2.f * (float)mx;
      out[2*(size_t)SOUT + v]     = dv;
      out[2*(size_t)SOUT + 4 + v] = dv;
    }
  }
}

// ---------------------------------------------------------------------------
// Fallback branches (dispatched on out_size).
// ---------------------------------------------------------------------------
__global__ void k_simple(const float* __restrict__ ws,
                         const int* __restrict__ udeg, const int* __restrict__ odeg,
                         float* __restrict__ out) {  // lf >= 0 branch
  int t = blockIdx.x * blockDim.x + threadIdx.x;
  if (t < NROWS) {
    for (int v = 0; v < 4; ++v)
      for (int i = 0; i < 4; ++i) {
        out[(t*4+v)*4 + i]         = ws[WS_U + (v*4+i)*TPAD + t];
        out[16400 + (t*4+v)*4 + i] = ws[WS_O + (v*4+i)*TPAD + t];
      }
  } else if (t == NROWS) {
    for (int v = 0; v < 4; ++v) {
      int mx = -2147483647;
      for (int k = 0; k < 32; ++k) mx = imax(mx, imax(udeg[k*4+v], odeg[k*4+v]));
      float dv = (float)mx;
      out[32800 + v] = dv; out[32804 + v] = dv;
    }
  }
}

__global__ void k_zero(float* __restrict__ out, int n) {
  int i = blockIdx.x * blockDim.x + threadIdx.x;
  if (i < n) out[i] = 0.f;
}

// ---------------------------------------------------------------------------
extern "C" void kernel_launch(void* const* d_in, const int* in_sizes, int n_in,
                              void* d_out, int out_size, void* d_ws, size_t ws_size,
                              hipStream_t stream) {
  const float* iu   = (const float*)d_in[0];
  const float* io   = (const float*)d_in[1];
  const int*   udeg = (const int*)  d_in[2];
  const int*   odeg = (const int*)  d_in[3];
  const float* posw = (const float*)d_in[4];
  const float* negw = (const float*)d_in[5];
  const float* bias = (const float*)d_in[6];
  float* out = (float*)d_out;
  float* ws  = (float*)d_ws;
  (void)in_sizes; (void)n_in; (void)ws_size;

  if (out_size <= 16) {                      // uf <= 0 branch: all zeros
    k_zero<<<1, 64, 0, stream>>>(out, out_size);
    return;
  }
  k_prep<<<(TPAD + 255)/256, 256, 0, stream>>>(iu, io, posw, negw, bias, ws);
  if (out_size == 2*SOUT + 8) {              // quad branch (expected)
    k_reduce<<<1, 256, 0, stream>>>(ws);
    // interior: guard-free stores (64x64 tiles)
    k_pairs_t<false><<<dim3(NTILE-1, NTILE-1), 32, 0, stream>>>(ws, out, 0, 0);
    // edges: b-column strip (all 65 a-tiles, bTile=64) and a-row strip (aTile=64, bTiles 0..63)
    k_pairs_t<true ><<<dim3(NTILE, 1),        32, 0, stream>>>(ws, out, 0, NTILE-1);
    k_pairs_t<true ><<<dim3(1, NTILE-1),      32, 0, stream>>>(ws, out, NTILE-1, 0);
    k_tail<<<9, 128, 0, stream>>>(ws, udeg, odeg, out);
  } else {                                   // lf >= 0 branch: pass-through
    k_simple<<<9, 128, 0, stream>>>(ws, udeg, odeg, out);
  }
}